// Loss_71416716198003
// MI455X (gfx1250) — compile-verified
//
#include <hip/hip_runtime.h>
#include <hip/hip_bf16.h>

typedef float v8f __attribute__((ext_vector_type(8)));
typedef float v2f __attribute__((ext_vector_type(2)));

#define NC 20000
#define NB 6890
#define NF 40000
#define NE 60000
#define NTILES 432              /* ceil(6890/16)=431, padded to even 432 */
#define NBP_ALLOC ((NTILES + 1) * 16)  /* 6928: +1 dummy tile for prefetch overrun */

#define T_STEP (1.0f / 30.0f)
#define DENSITY 426.0f
#define THICK 0.00047f
#define G_ACC 9.81f
#define LAM 20.9f
#define MU_C 11.1f
#define EPS_COLL 0.002f
#define K_COLL 250.0f

/* workspace layout in floats */
#define OFF_PACK 0                       /* NBP_ALLOC*4 = 27712 */
#define OFF_M    27712                   /* NC */
#define OFF_IDX  47712                   /* NC (int) */
#define OFF_N    67712                   /* NF*3 = 120000 */
#define OFF_PV   187712                  /* 79 vertex partials */
#define OFF_PS   187791                  /* 157 strain partials */
#define OFF_PB   187948                  /* 235 bending partials */
#define OFF_PD   188183                  /* 79 collision-dot partials */
#define OFF_PC   188262                  /* 79 collision-count partials */

#define NBLK_V 79    /* ceil(20000/256) */
#define NBLK_F 157   /* ceil(40000/256) */
#define NBLK_E 235   /* ceil(60000/256) */

__device__ __forceinline__ float blockReduceSum256(float v) {
    __shared__ float s[256];
    int t = threadIdx.x;
    __syncthreads();
    s[t] = v;
    __syncthreads();
    for (int off = 128; off > 0; off >>= 1) {
        if (t < off) s[t] += s[t + off];
        __syncthreads();
    }
    return s[0];
}

/* ---------- 1. pack body verts into (-2b, |b|^2), padded ---------- */
__global__ void pack_body_kernel(const float* __restrict__ vb, float* __restrict__ pack) {
    int j = blockIdx.x * blockDim.x + threadIdx.x;
    if (j >= NBP_ALLOC) return;
    float4 p;
    if (j < NB) {
        float bx = vb[j * 3 + 0], by = vb[j * 3 + 1], bz = vb[j * 3 + 2];
        p = make_float4(-2.0f * bx, -2.0f * by, -2.0f * bz, bx * bx + by * by + bz * bz);
    } else {
        p = make_float4(0.0f, 0.0f, 0.0f, 3.0e37f);  /* padding never wins the min */
    }
    ((float4*)pack)[j] = p;
}

/* ---------- 2. normalized face normals: N = cross(x2-x0, x1-x0)/|.| ---------- */
__global__ void normals_kernel(const float* __restrict__ x, const int* __restrict__ f,
                               float* __restrict__ Nout) {
    int i = blockIdx.x * blockDim.x + threadIdx.x;
    if (i >= NF) return;
    int i0 = f[i * 3 + 0], i1 = f[i * 3 + 1], i2 = f[i * 3 + 2];
    float ax = x[i0 * 3], ay = x[i0 * 3 + 1], az = x[i0 * 3 + 2];
    float bx = x[i1 * 3], by = x[i1 * 3 + 1], bz = x[i1 * 3 + 2];
    float cx = x[i2 * 3], cy = x[i2 * 3 + 1], cz = x[i2 * 3 + 2];
    float ux = cx - ax, uy = cy - ay, uz = cz - az;  /* V02 */
    float vx = bx - ax, vy = by - ay, vz = bz - az;  /* V01 */
    float nx = uy * vz - uz * vy;
    float ny = uz * vx - ux * vz;
    float nz = ux * vy - uy * vx;
    float inv = rsqrtf(nx * nx + ny * ny + nz * nz);
    Nout[i * 3 + 0] = nx * inv;
    Nout[i * 3 + 1] = ny * inv;
    Nout[i * 3 + 2] = nz * inv;
}

/* ---------- 3. inertia + gravity per-vertex partials ---------- */
__global__ void vert_kernel(const float* __restrict__ x1, const float* __restrict__ x2,
                            const float* __restrict__ x3, const float* __restrict__ surfp,
                            float* __restrict__ partial) {
    int i = blockIdx.x * blockDim.x + threadIdx.x;
    float contrib = 0.0f;
    if (i < NC) {
        const float t = T_STEP;
        float vm = surfp[0] * DENSITY * THICK / (float)NC;
        float a1x = x1[i * 3], a1y = x1[i * 3 + 1], a1z = x1[i * 3 + 2];
        float a2x = x2[i * 3], a2y = x2[i * 3 + 1], a2z = x2[i * 3 + 2];
        float a3x = x3[i * 3], a3y = x3[i * 3 + 1], a3z = x3[i * 3 + 2];
        float hx = a2x + (a2x - a1x) / t * t;
        float hy = a2y + (a2y - a1y) / t * t;
        float hz = a2z + (a2z - a1z) / t * t;
        float dx = a3x - hx, dy = a3y - hy, dz = a3z - hz;
        float xb = dx * dx + dy * dy + dz * dz;
        contrib = xb * vm / (2.0f * t * t) + vm * G_ACC * a3z;
    }
    float s = blockReduceSum256(contrib);
    if (threadIdx.x == 0) partial[blockIdx.x] = s;
}

/* ---------- 4. strain energy partials ---------- */
__global__ void strain_kernel(const float* __restrict__ Xr, const float* __restrict__ x,
                              const int* __restrict__ f, float* __restrict__ partial) {
    int i = blockIdx.x * blockDim.x + threadIdx.x;
    float contrib = 0.0f;
    if (i < NF) {
        int i0 = f[i * 3 + 0], i1 = f[i * 3 + 1], i2 = f[i * 3 + 2];
        float x0[3], x1c[3], x2c[3], X0[3], X1[3], X2[3];
        #pragma unroll
        for (int k = 0; k < 3; ++k) {
            x0[k] = x[i0 * 3 + k]; x1c[k] = x[i1 * 3 + k]; x2c[k] = x[i2 * 3 + k];
            X0[k] = Xr[i0 * 3 + k]; X1[k] = Xr[i1 * 3 + k]; X2[k] = Xr[i2 * 3 + k];
        }
        /* area from deformed triangle */
        float ux = x2c[0] - x0[0], uy = x2c[1] - x0[1], uz = x2c[2] - x0[2];
        float vx = x1c[0] - x0[0], vy = x1c[1] - x0[1], vz = x1c[2] - x0[2];
        float crx = uy * vz - uz * vy, cry = uz * vx - ux * vz, crz = ux * vy - uy * vx;
        float area = 0.5f * sqrtf(crx * crx + cry * cry + crz * crz);
        /* inverse of X_M (columns = X0,X1,X2) via adjugate */
        float r0[3], r1[3], r2[3];
        r0[0] = X1[1] * X2[2] - X1[2] * X2[1];   /* c1 x c2 */
        r0[1] = X1[2] * X2[0] - X1[0] * X2[2];
        r0[2] = X1[0] * X2[1] - X1[1] * X2[0];
        r1[0] = X2[1] * X0[2] - X2[2] * X0[1];   /* c2 x c0 */
        r1[1] = X2[2] * X0[0] - X2[0] * X0[2];
        r1[2] = X2[0] * X0[1] - X2[1] * X0[0];
        r2[0] = X0[1] * X1[2] - X0[2] * X1[1];   /* c0 x c1 */
        r2[1] = X0[2] * X1[0] - X0[0] * X1[2];
        r2[2] = X0[0] * X1[1] - X0[1] * X1[0];
        float det = X0[0] * r0[0] + X0[1] * r0[1] + X0[2] * r0[2];
        float invd = 1.0f / det;
        /* Fm[a][b] = x0_a*inv0b + x1_a*inv1b + x2_a*inv2b ; inv rows = r0,r1,r2 * invd */
        float Fm[3][3];
        #pragma unroll
        for (int a = 0; a < 3; ++a) {
            #pragma unroll
            for (int b = 0; b < 3; ++b) {
                Fm[a][b] = (x0[a] * r0[b] + x1c[a] * r1[b] + x2c[a] * r2[b]) * invd;
            }
        }
        /* G = 0.5 (Fm^T Fm - I), symmetric */
        float C[3][3];
        #pragma unroll
        for (int a = 0; a < 3; ++a) {
            #pragma unroll
            for (int b = 0; b < 3; ++b) {
                C[a][b] = Fm[0][a] * Fm[0][b] + Fm[1][a] * Fm[1][b] + Fm[2][a] * Fm[2][b];
            }
        }
        float trG = 0.5f * (C[0][0] + C[1][1] + C[2][2] - 3.0f);
        float trGG = 0.0f;
        #pragma unroll
        for (int a = 0; a < 3; ++a) {
            #pragma unroll
            for (int b = 0; b < 3; ++b) {
                float g = 0.5f * (C[a][b] - (a == b ? 1.0f : 0.0f));
                trGG += g * g;
            }
        }
        float phi = (LAM * 0.5f) * trG * trG + MU_C * trGG;
        contrib = THICK * area * phi;
    }
    float s = blockReduceSum256(contrib);
    if (threadIdx.x == 0) partial[blockIdx.x] = s;
}

/* ---------- 5. bending partials: sum(1 - dot(N0,N1)) ---------- */
__global__ void bend_kernel(const float* __restrict__ Nf, const int* __restrict__ neighF,
                            float* __restrict__ partial) {
    int i = blockIdx.x * blockDim.x + threadIdx.x;
    float contrib = 0.0f;
    if (i < NE) {
        int fa = neighF[i * 2 + 0], fb = neighF[i * 2 + 1];
        float d = Nf[fa * 3] * Nf[fb * 3] + Nf[fa * 3 + 1] * Nf[fb * 3 + 1] +
                  Nf[fa * 3 + 2] * Nf[fb * 3 + 2];
        contrib = 1.0f - d;
    }
    float s = blockReduceSum256(contrib);
    if (threadIdx.x == 0) partial[blockIdx.x] = s;
}

/* ---------- 6. WMMA nearest-body search ----------
 * Per wave: 16 cloth rows. m_i = min_j (-2 c_i . b_j + |b_j|^2) via
 * V_WMMA_F32_16X16X4_F32 tiles: A row = (cx,cy,cz,1), B col = (-2b, |b|^2).
 * A 16x4 layout: lanes0-15 VGPR{0,1}=K{0,1}; lanes16-31 VGPR{0,1}=K{2,3}.
 * B 4x16 mirrors with N across lanes. D VGPR r: lanes0-15 row r, lanes16-31 row r+8.
 * B tile loads are software-pipelined (rotating cur/nxt, one dummy tile of
 * padding at the end of the packed table so the prefetch never goes OOB).
 */
__global__ __launch_bounds__(32) void coll_min_kernel(const float* __restrict__ v3,
                                                      const float* __restrict__ bodyPack,
                                                      float* __restrict__ mOut,
                                                      int* __restrict__ idxOut) {
    int lane = threadIdx.x;
    int half = (lane < 16) ? 1 : 0;
    int m = lane & 15;
    int rowBase = blockIdx.x * 16;           /* NC % 16 == 0 */
    int row = rowBase + m;
    float cx = v3[row * 3 + 0], cy = v3[row * 3 + 1], cz = v3[row * 3 + 2];
    v2f a;
    a.x = half ? cx : cz;
    a.y = half ? cy : 1.0f;

    float minv[8];
    int mini[8];
    #pragma unroll
    for (int r = 0; r < 8; ++r) { minv[r] = 3.4e38f; mini[r] = 0; }

    const float4* bp4 = (const float4*)bodyPack;
    float4 cur = bp4[m];                     /* tile 0 */
    #pragma unroll 2
    for (int t = 0; t < NTILES; ++t) {
        float4 nxt = bp4[(t + 1) * 16 + m];  /* prefetch next tile (padded) */
        v2f b;
        b.x = half ? cur.x : cur.z;
        b.y = half ? cur.y : cur.w;
        v8f c = {};
        c = __builtin_amdgcn_wmma_f32_16x16x4_f32(false, a, false, b, (short)0, c,
                                                  false, false);
        int j = t * 16 + m;
        #pragma unroll
        for (int r = 0; r < 8; ++r) {
            float v = c[r];
            if (v < minv[r]) { minv[r] = v; mini[r] = j; }  /* strict < keeps first min */
        }
        cur = nxt;
    }
    /* reduce across the 16 lanes of each half (xor masks stay within halves) */
    #pragma unroll
    for (int off = 8; off >= 1; off >>= 1) {
        #pragma unroll
        for (int r = 0; r < 8; ++r) {
            float ov = __shfl_xor(minv[r], off, 32);
            int oi = __shfl_xor(mini[r], off, 32);
            if (ov < minv[r] || (ov == minv[r] && oi < mini[r])) {
                minv[r] = ov; mini[r] = oi;
            }
        }
    }
    if (m == 0) {
        int rOff = half ? 0 : 8;
        #pragma unroll
        for (int r = 0; r < 8; ++r) {
            mOut[rowBase + rOff + r] = minv[r];
            idxOut[rowBase + rOff + r] = mini[r];
        }
    }
}

/* ---------- 7. collision finalize: masked dot + count partials ---------- */
__global__ void coll_fin_kernel(const float* __restrict__ v3, const float* __restrict__ vb,
                                const float* __restrict__ vn, const float* __restrict__ mIn,
                                const int* __restrict__ idxIn, float* __restrict__ pDot,
                                float* __restrict__ pCnt) {
    int i = blockIdx.x * blockDim.x + threadIdx.x;
    float dsum = 0.0f, csum = 0.0f;
    if (i < NC) {
        float cx = v3[i * 3], cy = v3[i * 3 + 1], cz = v3[i * 3 + 2];
        float sx = cx * cx + cy * cy + cz * cz;
        float dist = sqrtf(sx + mIn[i] + 1e-10f);
        if (dist < EPS_COLL) {
            int j = idxIn[i];
            float dot = (vb[j * 3] - cx) * vn[j * 3] +
                        (vb[j * 3 + 1] - cy) * vn[j * 3 + 1] +
                        (vb[j * 3 + 2] - cz) * vn[j * 3 + 2];
            if (dot > 0.0f) { dsum = dot; csum = 1.0f; }
        }
    }
    float s1 = blockReduceSum256(dsum);
    float s2 = blockReduceSum256(csum);
    if (threadIdx.x == 0) { pDot[blockIdx.x] = s1; pCnt[blockIdx.x] = s2; }
}

/* ---------- 8. deterministic scalar combine ---------- */
__global__ void combine_kernel(const float* __restrict__ pv, const float* __restrict__ ps,
                               const float* __restrict__ pb, const float* __restrict__ pd,
                               const float* __restrict__ pc, float* __restrict__ out) {
    if (threadIdx.x != 0 || blockIdx.x != 0) return;
    float s_ig = 0.0f;
    for (int k = 0; k < NBLK_V; ++k) s_ig += pv[k];
    float s_st = 0.0f;
    for (int k = 0; k < NBLK_F; ++k) s_st += ps[k];
    float s_bd = 0.0f;
    for (int k = 0; k < NBLK_E; ++k) s_bd += pb[k];
    float s_dot = 0.0f, s_cnt = 0.0f;
    for (int k = 0; k < NBLK_V; ++k) { s_dot += pd[k]; s_cnt += pc[k]; }
    float cnt = fmaxf(s_cnt, 1.0f);
    out[0] = s_ig + s_st / (float)NF + s_bd / (float)NE + K_COLL * s_dot / cnt;
}

extern "C" void kernel_launch(void* const* d_in, const int* in_sizes, int n_in,
                              void* d_out, int out_size, void* d_ws, size_t ws_size,
                              hipStream_t stream) {
    const float* T_cloth = (const float*)d_in[0];
    const float* v1 = (const float*)d_in[1];
    const float* v2 = (const float*)d_in[2];
    const float* v3 = (const float*)d_in[3];
    const float* v_body = (const float*)d_in[4];
    const float* v_bn = (const float*)d_in[5];
    const int* f_cloth = (const int*)d_in[6];
    const int* neighF = (const int*)d_in[7];
    const float* surface = (const float*)d_in[8];
    float* out = (float*)d_out;

    float* ws = (float*)d_ws;
    float* pack = ws + OFF_PACK;
    float* mBuf = ws + OFF_M;
    int* idxBuf = (int*)(ws + OFF_IDX);
    float* Nbuf = ws + OFF_N;
    float* pv = ws + OFF_PV;
    float* ps = ws + OFF_PS;
    float* pb = ws + OFF_PB;
    float* pd = ws + OFF_PD;
    float* pc = ws + OFF_PC;

    pack_body_kernel<<<(NBP_ALLOC + 255) / 256, 256, 0, stream>>>(v_body, pack);
    normals_kernel<<<NBLK_F, 256, 0, stream>>>(v3, f_cloth, Nbuf);
    vert_kernel<<<NBLK_V, 256, 0, stream>>>(v1, v2, v3, surface, pv);
    strain_kernel<<<NBLK_F, 256, 0, stream>>>(T_cloth, v3, f_cloth, ps);
    bend_kernel<<<NBLK_E, 256, 0, stream>>>(Nbuf, neighF, pb);
    coll_min_kernel<<<NC / 16, 32, 0, stream>>>(v3, pack, mBuf, idxBuf);
    coll_fin_kernel<<<NBLK_V, 256, 0, stream>>>(v3, v_body, v_bn, mBuf, idxBuf, pd, pc);
    combine_kernel<<<1, 1, 0, stream>>>(pv, ps, pb, pd, pc, out);
}